// StackLSTMCell_52836687675633
// MI455X (gfx1250) — compile-verified
//
#include <hip/hip_runtime.h>
#include <math.h>

// ---------------------------------------------------------------------------
// StackLSTMCell for MI455X (gfx1250): bandwidth-bound (135MB stack copy-out)
// + fp32 WMMA (V_WMMA_F32_16X16X4_F32) for the 4 fused-gate GEMMs.
// ---------------------------------------------------------------------------

typedef __attribute__((ext_vector_type(2))) float v2f;
typedef __attribute__((ext_vector_type(8))) float v8f;

#define BB   256          // batch
#define INN  256          // input dim
#define HH   256          // hidden dim
#define LL   2            // layers
#define SS   128          // stack depth
#define G4H  1024         // 4*H
#define KD   256          // GEMM K per operand (IN == H == 256)
#define STACK_ELEMS ((long)(SS + 1) * BB * HH * LL)   // 16,908,288 floats

// workspace layout (float offsets)
#define WS_CURH0   0
#define WS_CURH1   65536
#define WS_CURC0   131072
#define WS_CURC1   196608
#define WS_GATES   262144      // B*4H = 262144 floats (reused by both layers)
#define WS_H0      524288
#define WS_C0      589824
#define WS_H1      655360
#define WS_C1      720896

__device__ __forceinline__ float sigm_(float x) { return 1.0f / (1.0f + expf(-x)); }

// ---------------------------------------------------------------------------
// Gather top-of-stack state (both layers) into contiguous per-layer buffers.
// Layer dim is innermost in the stacks -> one float2 load covers both layers.
// ---------------------------------------------------------------------------
__global__ __launch_bounds__(256) void stacklstm_gather(
    const float* __restrict__ hstack, const float* __restrict__ cstack,
    const int* __restrict__ pos, float* __restrict__ ws)
{
    int t = blockIdx.x * blockDim.x + threadIdx.x;   // 0 .. B*H-1
    int b = t >> 8;
    int h = t & 255;
    long base = ((long)pos[b] * BB + b) * (HH * LL) + (long)h * LL;
    float2 hv = *(const float2*)(hstack + base);
    float2 cv = *(const float2*)(cstack + base);
    ws[WS_CURH0 + t] = hv.x;
    ws[WS_CURH1 + t] = hv.y;
    ws[WS_CURC0 + t] = cv.x;
    ws[WS_CURC1 + t] = cv.y;
}

// ---------------------------------------------------------------------------
// gates(B x 4H) = X(B x 256) @ Wih^T + Hprev(B x 256) @ Whh^T + bih + bhh
// One wave per 16x16 output tile, fp32 WMMA 16x16x4, fused K = 512.
//
// Lane fragment addressing (wave32):
//   A (16x4 f32):  row M = lane&15, VGPR0/1 hold K = 2*(lane>>4) + {0,1}
//   B (4x16 f32):  col N = lane&15, VGPR0/1 hold K = 2*(lane>>4) + {0,1}
//     (B tile element [k][n] = W[(n0+n)*256 + k], i.e. W^T)
//   C/D (16x16):   acc[v] = (M = v + 8*(lane>>4), N = lane&15)
// Both A and B fragments are 2 consecutive floats -> b64 loads.
// ---------------------------------------------------------------------------
__global__ __launch_bounds__(128) void stacklstm_gemm(
    const float* __restrict__ X,   const float* __restrict__ Hp,
    const float* __restrict__ Wih, const float* __restrict__ Whh,
    const float* __restrict__ bih, const float* __restrict__ bhh,
    float* __restrict__ gates)
{
    const int wave = threadIdx.x >> 5;
    const int lane = threadIdx.x & 31;
    const int tile = blockIdx.x * 4 + wave;      // 0..1023
    const int m0 = (tile >> 6) << 4;             // 16 row-tiles  (B/16)
    const int n0 = (tile & 63) << 4;             // 64 col-tiles  (4H/16)
    const int lr = lane & 15;                    // M (A) / N (B) within tile
    const int lk = (lane >> 4) << 1;             // K sub-offset {0,2}

    const float* ap  = X   + (m0 + lr) * KD + lk;
    const float* hpp = Hp  + (m0 + lr) * KD + lk;
    const float* wp1 = Wih + (n0 + lr) * KD + lk;
    const float* wp2 = Whh + (n0 + lr) * KD + lk;

    const float bias = bih[n0 + lr] + bhh[n0 + lr];   // depends on N only
    v8f acc = {bias, bias, bias, bias, bias, bias, bias, bias};

#pragma unroll 8
    for (int k = 0; k < KD; k += 4) {
        v2f a = *(const v2f*)(ap + k);
        v2f w = *(const v2f*)(wp1 + k);
        acc = __builtin_amdgcn_wmma_f32_16x16x4_f32(
            false, a, false, w, (short)0, acc, false, false);
    }
#pragma unroll 8
    for (int k = 0; k < KD; k += 4) {
        v2f a = *(const v2f*)(hpp + k);
        v2f w = *(const v2f*)(wp2 + k);
        acc = __builtin_amdgcn_wmma_f32_16x16x4_f32(
            false, a, false, w, (short)0, acc, false, false);
    }

    const int nn = n0 + lr;
    const int mb = m0 + ((lane >> 4) << 3);
#pragma unroll
    for (int v = 0; v < 8; ++v)
        gates[(long)(mb + v) * G4H + nn] = acc[v];
}

// ---------------------------------------------------------------------------
// Gate nonlinearities (torch LSTMCell order: i, f, g, o).
// ---------------------------------------------------------------------------
__global__ __launch_bounds__(256) void stacklstm_act(
    const float* __restrict__ gates, const float* __restrict__ cprev,
    float* __restrict__ hnew, float* __restrict__ cnew)
{
    int t = blockIdx.x * blockDim.x + threadIdx.x;   // 0 .. B*H-1
    int b = t >> 8;
    int h = t & 255;
    const float* g = gates + (long)b * G4H;
    float gi = g[h];
    float gf = g[HH + h];
    float gg = g[2 * HH + h];
    float go = g[3 * HH + h];
    float c2 = sigm_(gf) * cprev[t] + sigm_(gi) * tanhf(gg);
    float h2 = sigm_(go) * tanhf(c2);
    hnew[t] = h2;
    cnew[t] = c2;
}

// ---------------------------------------------------------------------------
// Scatter new state at pos+1 into the (already copied) output stacks, and
// emit h_out/c_out = returned state at new_pos = pos+op, H row 255, per layer.
// For op==+1 new_pos == pos+1 (same b) -> use freshly computed values.
// ---------------------------------------------------------------------------
__global__ __launch_bounds__(256) void stacklstm_finish(
    const float* __restrict__ h0, const float* __restrict__ c0,
    const float* __restrict__ h1, const float* __restrict__ c1,
    const float* __restrict__ hin, const float* __restrict__ cin,
    const int* __restrict__ op, const int* __restrict__ pos,
    float* __restrict__ out)
{
    int t = blockIdx.x * blockDim.x + threadIdx.x;   // 0 .. B*H-1
    int b = t >> 8;
    int h = t & 255;

    float* hstack_out = out + 2 * BB * LL;                 // after h_out,c_out
    float* cstack_out = hstack_out + STACK_ELEMS;

    long base = ((long)(pos[b] + 1) * BB + b) * (HH * LL) + (long)h * LL;
    float2 nh = make_float2(h0[t], h1[t]);
    float2 nc = make_float2(c0[t], c1[t]);
    *(float2*)(hstack_out + base) = nh;
    *(float2*)(cstack_out + base) = nc;

    if (h == HH - 1) {
        int o = op[b];
        int np = pos[b] + o;
#pragma unroll
        for (int l = 0; l < LL; ++l) {
            float hv, cv;
            if (o == 1) {                       // reads the row we just pushed
                hv = (l == 0) ? h0[t] : h1[t];
                cv = (l == 0) ? c0[t] : c1[t];
            } else {                            // old stack content
                long rb = ((long)np * BB + b) * (HH * LL) + (long)(HH - 1) * LL + l;
                hv = hin[rb];
                cv = cin[rb];
            }
            out[b * LL + l] = hv;               // h_out (1,B,L)
            out[BB * LL + b * LL + l] = cv;     // c_out (1,B,L)
        }
    }
}

// ---------------------------------------------------------------------------
extern "C" void kernel_launch(void* const* d_in, const int* in_sizes, int n_in,
                              void* d_out, int out_size, void* d_ws, size_t ws_size,
                              hipStream_t stream)
{
    const float* x     = (const float*)d_in[0];
    const float* hst   = (const float*)d_in[1];
    const float* cst   = (const float*)d_in[2];
    const float* wih0  = (const float*)d_in[3];
    const float* whh0  = (const float*)d_in[4];
    const float* bih0  = (const float*)d_in[5];
    const float* bhh0  = (const float*)d_in[6];
    const float* wih1  = (const float*)d_in[7];
    const float* whh1  = (const float*)d_in[8];
    const float* bih1  = (const float*)d_in[9];
    const float* bhh1  = (const float*)d_in[10];
    const int*   op    = (const int*)d_in[11];
    const int*   pos   = (const int*)d_in[12];

    float* out = (float*)d_out;
    float* ws  = (float*)d_ws;

    float* hstack_out = out + 2 * BB * LL;
    float* cstack_out = hstack_out + STACK_ELEMS;

    // 1) bulk copy stacks to output (dominant cost: ~270MB HBM traffic)
    hipMemcpyAsync(hstack_out, hst, STACK_ELEMS * sizeof(float),
                   hipMemcpyDeviceToDevice, stream);
    hipMemcpyAsync(cstack_out, cst, STACK_ELEMS * sizeof(float),
                   hipMemcpyDeviceToDevice, stream);

    // 2) gather top-of-stack state
    stacklstm_gather<<<BB * HH / 256, 256, 0, stream>>>(hst, cst, pos, ws);

    // 3) layer 0: gates -> activations
    stacklstm_gemm<<<256, 128, 0, stream>>>(
        x, ws + WS_CURH0, wih0, whh0, bih0, bhh0, ws + WS_GATES);
    stacklstm_act<<<BB * HH / 256, 256, 0, stream>>>(
        ws + WS_GATES, ws + WS_CURC0, ws + WS_H0, ws + WS_C0);

    // 4) layer 1 (input = layer-0 hidden)
    stacklstm_gemm<<<256, 128, 0, stream>>>(
        ws + WS_H0, ws + WS_CURH1, wih1, whh1, bih1, bhh1, ws + WS_GATES);
    stacklstm_act<<<BB * HH / 256, 256, 0, stream>>>(
        ws + WS_GATES, ws + WS_CURC1, ws + WS_H1, ws + WS_C1);

    // 5) scatter at pos+1, gather return state at pos+op
    stacklstm_finish<<<BB * HH / 256, 256, 0, stream>>>(
        ws + WS_H0, ws + WS_C0, ws + WS_H1, ws + WS_C1,
        hst, cst, op, pos, out);
}